// SelectiveSSMBlock_63651415327124
// MI455X (gfx1250) — compile-verified
//
#include <hip/hip_runtime.h>
#include <hip/hip_bf16.h>

// Shapes (compile-time constants from the reference)
#define BATCH 2
#define PC    48
#define IMG   128
#define WIN   2
#define NS    64           // IMG/WIN
#define L     4096         // NS*NS
#define DDIM  192          // PC*WIN*WIN (token dim == state dim)

typedef __attribute__((ext_vector_type(2))) float v2f;
typedef __attribute__((ext_vector_type(8))) float v8f;

__device__ __forceinline__ float softplus_f(float x) {
    // numerically stable log1p(exp(x))
    return fmaxf(x, 0.0f) + log1pf(__expf(-fabsf(x)));
}

// ---------------------------------------------------------------------------
// Kernel 1: window-unfold x (B,PC,IMG,IMG) -> tokens (B,L,D), channel-major D
// tokens[b, l, d] with d = c*4 + wy*2 + wx, l = by*NS + bx
// ---------------------------------------------------------------------------
__global__ void pack_tokens(const float* __restrict__ x, float* __restrict__ tok) {
    int idx = blockIdx.x * blockDim.x + threadIdx.x;     // 0 .. B*L*D-1 (exact grid)
    int d  = idx % DDIM;
    int bl = idx / DDIM;
    int l  = bl % L;
    int b  = bl / L;
    int c  = d >> 2;
    int wy = (d >> 1) & 1;
    int wx = d & 1;
    int by = l >> 6;           // l / NS
    int bx = l & 63;           // l % NS
    size_t src = (((size_t)b * PC + c) * IMG + (by * WIN + wy)) * IMG + (bx * WIN + wx);
    tok[idx] = x[src];
}

// ---------------------------------------------------------------------------
// Kernel 2: projections out = tok @ W^T + bias  (M=B*L=8192, N=K=192)
// One wave32 computes a 16x16 tile with V_WMMA_F32_16X16X4_F32, K-loop of 48.
// blockIdx.z selects {0:B, 1:C, 2:dt(double-softplus)}.
// ---------------------------------------------------------------------------
__global__ void proj_wmma(const float* __restrict__ tok,
                          const float* __restrict__ W_B, const float* __restrict__ b_B,
                          const float* __restrict__ W_C, const float* __restrict__ b_C,
                          const float* __restrict__ W_dt, const float* __restrict__ b_dt,
                          float* __restrict__ Bm, float* __restrict__ Cm,
                          float* __restrict__ Dl) {
    const int mode = blockIdx.z;
    const float* __restrict__ W    = (mode == 0) ? W_B : (mode == 1) ? W_C : W_dt;
    const float* __restrict__ bias = (mode == 0) ? b_B : (mode == 1) ? b_C : b_dt;
    float* __restrict__ out        = (mode == 0) ? Bm  : (mode == 1) ? Cm  : Dl;

    const int row0 = blockIdx.x * 16;                       // M tile (512 tiles)
    const int col0 = (blockIdx.y * 4 + threadIdx.y) * 16;   // N tile (12 tiles)
    const int lane = threadIdx.x;                           // wave32
    const int half = lane >> 4;                             // 0 or 1
    const int m16  = lane & 15;

    // A frag: lane holds tok[row0+m16, k0 + 2*half + {0,1}]
    // B frag: lane holds W[col0+m16,  k0 + 2*half + {0,1}]  (B = W^T, row-major W)
    const float* __restrict__ arow = tok + (size_t)(row0 + m16) * DDIM + 2 * half;
    const float* __restrict__ brow = W   + (size_t)(col0 + m16) * DDIM + 2 * half;

    v8f acc = {};
#pragma unroll 8
    for (int k0 = 0; k0 < DDIM; k0 += 4) {
        v2f a = *(const v2f*)(arow + k0);
        v2f b = *(const v2f*)(brow + k0);
        acc = __builtin_amdgcn_wmma_f32_16x16x4_f32(
            /*neg_a=*/false, a, /*neg_b=*/false, b,
            /*c_mod=*/(short)0, acc, /*reuse_a=*/false, /*reuse_b=*/false);
    }

    const int ncol = col0 + m16;
    const float bv = bias[ncol];
    const int rbase = row0 + 8 * half;
#pragma unroll
    for (int r = 0; r < 8; ++r) {
        float v = acc[r] + bv;                 // linear + bias
        if (mode == 2) {
            // delta = softplus(softplus(lin) + b_dt)
            v = softplus_f(softplus_f(v) + bv);
        }
        out[(size_t)(rbase + r) * DDIM + ncol] = v;
    }
}

// ---------------------------------------------------------------------------
// Kernel 3: selective scan. One wave32 per (b,d) recurrence; state h[d, n]
// striped 6 values/lane over n. Sequential over L=4096. Output scatter fused.
// ---------------------------------------------------------------------------
__global__ void ssm_scan(const float* __restrict__ tok,
                         const float* __restrict__ Bm,
                         const float* __restrict__ Cm,
                         const float* __restrict__ Dl,
                         const float* __restrict__ A_log,
                         const float* __restrict__ D_param,
                         float* __restrict__ out) {
    const int lane = threadIdx.x;           // 0..31
    const int bidx = blockIdx.x;            // 0 .. BATCH*DDIM-1
    const int b = bidx / DDIM;
    const int d = bidx % DDIM;

    // A[d, n] = -exp(A_log[d, n]); n = lane + 32*k
    float aneg[6], h[6];
    const float* __restrict__ Arow = A_log + (size_t)d * DDIM;
#pragma unroll
    for (int k = 0; k < 6; ++k) {
        aneg[k] = -__expf(Arow[lane + 32 * k]);
        h[k] = 0.0f;
    }
    const float Dp = D_param[d];

    const size_t plane = (size_t)b * L * DDIM;
    const float* __restrict__ Bp = Bm  + plane;
    const float* __restrict__ Cp = Cm  + plane;
    const float* __restrict__ dp = Dl  + plane + d;
    const float* __restrict__ up = tok + plane + d;

    // output scatter coords for this d
    const int c  = d >> 2;
    const int wy = (d >> 1) & 1;
    const int wx = d & 1;
    float* __restrict__ outb = out + (size_t)b * PC * IMG * IMG + (size_t)c * IMG * IMG;

    const float gate = 0.7310585786300049f;     // silu(1) = sigmoid(1)

    for (int l0 = 0; l0 < L; l0 += 32) {
        // batch-load 32 steps of the per-(l,d) scalars, broadcast below
        float dv = dp[(size_t)(l0 + lane) * DDIM];
        float uv = up[(size_t)(l0 + lane) * DDIM];
#pragma unroll 4
        for (int j = 0; j < 32; ++j) {
            const int l = l0 + j;
            const float* __restrict__ Brow = Bp + (size_t)l * DDIM + lane;
            const float* __restrict__ Crow = Cp + (size_t)l * DDIM + lane;
            if (l + 8 < L) {
                // cover the full 768B row: lane*6 floats spans all cachelines
                __builtin_prefetch(Bp + (size_t)(l + 8) * DDIM + lane * 6, 0, 0);
                __builtin_prefetch(Cp + (size_t)(l + 8) * DDIM + lane * 6, 0, 0);
            }
            const float dl = __shfl(dv, j, 32);
            const float ul = __shfl(uv, j, 32);
            const float du = dl * ul;
            float yp = 0.0f;
#pragma unroll
            for (int k = 0; k < 6; ++k) {
                float bn = Brow[32 * k];
                float cn = Crow[32 * k];
                h[k] = __expf(dl * aneg[k]) * h[k] + du * bn;
                yp = fmaf(h[k], cn, yp);
            }
            // wave32 butterfly reduction over n
#pragma unroll
            for (int off = 16; off > 0; off >>= 1)
                yp += __shfl_xor(yp, off, 32);
            if (lane == 0) {
                const int by = l >> 6;
                const int bx = l & 63;
                outb[(size_t)(by * WIN + wy) * IMG + (bx * WIN + wx)] =
                    (yp + Dp * ul) * gate;
            }
        }
    }
}

// ---------------------------------------------------------------------------
extern "C" void kernel_launch(void* const* d_in, const int* in_sizes, int n_in,
                              void* d_out, int out_size, void* d_ws, size_t ws_size,
                              hipStream_t stream) {
    const float* x       = (const float*)d_in[0];
    const float* A_log   = (const float*)d_in[1];
    const float* D_param = (const float*)d_in[2];
    const float* W_B     = (const float*)d_in[3];
    const float* b_B     = (const float*)d_in[4];
    const float* W_C     = (const float*)d_in[5];
    const float* b_C     = (const float*)d_in[6];
    const float* W_dt    = (const float*)d_in[7];
    const float* b_dt    = (const float*)d_in[8];
    float* out = (float*)d_out;

    const size_t P = (size_t)BATCH * L * DDIM;   // 1,572,864 floats per plane
    float* tok = (float*)d_ws;
    float* Bm  = tok + P;
    float* Cm  = Bm + P;
    float* Dl  = Cm + P;

    // 1) unfold into tokens
    pack_tokens<<<(P + 255) / 256, 256, 0, stream>>>(x, tok);

    // 2) three WMMA projections (+bias, dt gets double softplus)
    //    grid: 512 M-tiles x 3 (x4 waves = 12 N-tiles) x 3 matrices
    proj_wmma<<<dim3(512, 3, 3), dim3(32, 4), 0, stream>>>(
        tok, W_B, b_B, W_C, b_C, W_dt, b_dt, Bm, Cm, Dl);

    // 3) sequential selective scan, one wave per (b,d), fused output scatter
    ssm_scan<<<BATCH * DDIM, 32, 0, stream>>>(tok, Bm, Cm, Dl, A_log, D_param, out);
}